// CVaROptimizationLayer_26551487823918
// MI455X (gfx1250) — compile-verified
//
#include <hip/hip_runtime.h>
#include <math.h>

// CVaR portfolio optimization layer for gfx1250 (MI455X).
// One 256-thread (8x wave32) workgroup per batch element. scen (256x64 f32)
// lives in LDS with stride-65 rows; the 200-iteration subgradient loop runs
// fully on-chip. WMMA f32 16x16x4 for scenario GEMM and per-iter matvec,
// with loop-invariant WMMA A-fragments and gradient columns held in VGPRs.

typedef float v2f __attribute__((ext_vector_type(2)));
typedef float v8f __attribute__((ext_vector_type(8)));

#define NN 64
#define SS 256
#define BIL 4
#define Q_IDX 243
#define N_ITERS 200
#define INV_COEF 0.078125f /* 1/(S*(1-BETA)) = 1/12.8 */

// LDS layout (floats); 16B alignment of loss/coef/vv kept for b128 loads
#define OFF_SCEN 0          /* 256*65 = 16640 */
#define OFF_LS 16640        /* 64*65  = 4160  */
#define OFF_LOSS 20800      /* 256            */
#define OFF_COEF 21056      /* 256            */
#define OFF_W 21312         /* 64             */
#define OFF_V 21376         /* 64             */
#define OFF_U 21440         /* 64             */
#define OFF_GP 21504        /* 4*64           */
#define OFF_SCAL 21760      /* 16             */
#define LDS_FLOATS 21776

__device__ __forceinline__ void project_block(float* vv, float* uu, float* w,
                                              float* scal, int tid, float c) {
  const float z = 1.0f - c;
  // rank-based descending sort of 64 entries (stable tie-break on index)
  if (tid < NN) {
    float x = vv[tid];
    int r = 0;
    const float4* v4 = (const float4*)vv;
#pragma unroll 4
    for (int q = 0; q < 16; ++q) {
      float4 y = v4[q];
      int m = q * 4;
      r += ((y.x > x) || (y.x == x && (m + 0) < tid)) ? 1 : 0;
      r += ((y.y > x) || (y.y == x && (m + 1) < tid)) ? 1 : 0;
      r += ((y.z > x) || (y.z == x && (m + 2) < tid)) ? 1 : 0;
      r += ((y.w > x) || (y.w == x && (m + 3) < tid)) ? 1 : 0;
    }
    uu[r] = x;
  }
  __syncthreads();
  if (tid == 0) {
    // rho = count of (u_j * j - css_j) > 0, css = cumsum(u) - z  (1-indexed j)
    float cs = 0.0f;
    int rho = 0;
    for (int j = 0; j < NN; ++j) {
      cs += uu[j];
      if (uu[j] * (float)(j + 1) > (cs - z)) rho++;
    }
    float cs2 = 0.0f;
    for (int j = 0; j < rho; ++j) cs2 += uu[j];
    scal[0] = (cs2 - z) / (float)rho;  // theta
  }
  __syncthreads();
  if (tid < NN) {
    float t = vv[tid] - scal[0];
    float wn = (t > 0.0f) ? t : 0.0f;
    if (tid == BIL) wn += c;
    w[tid] = wn;
  }
  __syncthreads();
}

__global__ __launch_bounds__(256) void cvar_layer_kernel(
    const float* __restrict__ mu, const float* __restrict__ sigma,
    const float* __restrict__ eps, const float* __restrict__ is_crisis,
    float* __restrict__ out) {
  extern __shared__ float lds[];
  float* scen = lds + OFF_SCEN;  // [256][65]
  float* Ls = lds + OFF_LS;      // [64][65]
  float* loss = lds + OFF_LOSS;
  float* coef = lds + OFF_COEF;
  float* w = lds + OFF_W;
  float* vv = lds + OFF_V;
  float* uu = lds + OFF_U;
  float* gpart = lds + OFF_GP;
  float* scal = lds + OFF_SCAL;

  const int b = blockIdx.x;
  const int tid = threadIdx.x;
  const int lane = tid & 31;
  const int wv = tid >> 5;
  const int mrow = lane & 15;     // row/col within fragment
  const int halfsel = lane >> 4;  // 0: lanes 0-15, 1: lanes 16-31

  const float* epsB = eps + (size_t)b * SS * NN;
  const float* sigB = sigma + (size_t)b * NN * NN;
  const float* muB = mu + (size_t)b * NN;

  // Prefetch this block's eps tile into L2 while Cholesky runs
  __builtin_prefetch(epsB + tid * NN, 0, 1);

  // ---- 1) Async copy sigma -> LDS (CDNA5 GLOBAL_LOAD_ASYNC_TO_LDS path) ----
  {
#pragma unroll
    for (int r = 0; r < 16; ++r) {
      int e = tid + r * 256;
      int i = e >> 6, j = e & 63;
      unsigned ldsOff = (unsigned)((OFF_LS + i * 65 + j) * 4);
      const float* gp = sigB + e;
      asm volatile("global_load_async_to_lds_b32 %0, %1, off"
                   :
                   : "v"(ldsOff), "v"(gp)
                   : "memory");
    }
    asm volatile("s_wait_asynccnt 0" ::: "memory");
  }
  __syncthreads();
  // add 1e-6 jitter on diagonal
  if (tid < NN) Ls[tid * 65 + tid] += 1e-6f;
  __syncthreads();

  // ---- 2) In-LDS right-looking Cholesky (lower triangular) ----
  for (int k = 0; k < NN; ++k) {
    if (tid == 0) Ls[k * 65 + k] = sqrtf(Ls[k * 65 + k]);
    __syncthreads();
    if (tid > k && tid < NN) Ls[tid * 65 + k] /= Ls[k * 65 + k];
    __syncthreads();
    for (int e = tid; e < NN * NN; e += 256) {
      int i = e >> 6, j = e & 63;
      if (i > k && j > k && j <= i) Ls[i * 65 + j] -= Ls[i * 65 + k] * Ls[j * 65 + k];
    }
    __syncthreads();
  }
  // zero strict upper triangle so full-K GEMM is valid
  for (int e = tid; e < NN * NN; e += 256) {
    int i = e >> 6, j = e & 63;
    if (j > i) Ls[i * 65 + j] = 0.0f;
  }
  __syncthreads();

  // ---- 3) scen = mu + eps @ Ls^T via V_WMMA_F32_16X16X4_F32 ----
  // 64 tiles of 16(s) x 16(n); 8 tiles per wave. Fragments preloaded so the
  // 16 WMMAs per tile issue back-to-back instead of waiting per K-step.
  for (int t = 0; t < 8; ++t) {
    int tile = wv * 8 + t;
    int s0 = (tile >> 2) * 16;
    int n0 = (tile & 3) * 16;
    v2f ga[16], gb[16];
    const float* ap = epsB + (s0 + mrow) * NN + 2 * halfsel;  // A[m][k]=eps[s][k]
    const float* bp = Ls + (n0 + mrow) * 65 + 2 * halfsel;    // B[k][n]=L[n][k]
#pragma unroll
    for (int ks = 0; ks < 16; ++ks) {
      v2f ta = {ap[ks * 4], ap[ks * 4 + 1]};
      ga[ks] = ta;
    }
#pragma unroll
    for (int ks = 0; ks < 16; ++ks) {
      v2f tb = {bp[ks * 4], bp[ks * 4 + 1]};
      gb[ks] = tb;
    }
    v8f acc = {};
#pragma unroll
    for (int ks = 0; ks < 16; ++ks)
      acc = __builtin_amdgcn_wmma_f32_16x16x4_f32(false, ga[ks], false, gb[ks],
                                                  (short)0, acc, false, false);
    int colN = n0 + mrow;
    float m = muB[colN];
    int srow = s0 + 8 * halfsel;
#pragma unroll
    for (int v = 0; v < 8; ++v) scen[(srow + v) * 65 + colN] = acc[v] + m;
  }
  __syncthreads();

  // ---- 3b) Hoist loop-invariant fragments into registers ----
  // WMMA A-fragments of scen for the per-iteration matvec (2 s-tiles/wave)
  v2f aF[2][16];
#pragma unroll
  for (int half = 0; half < 2; ++half) {
    const float* rowp = scen + (wv * 32 + half * 16 + mrow) * 65 + 2 * halfsel;
#pragma unroll
    for (int ks = 0; ks < 16; ++ks) {
      v2f ta = {rowp[ks * 4], rowp[ks * 4 + 1]};
      aF[half][ks] = ta;
    }
  }
  // Gradient column of scen for this thread (chunk ch of 64 scenarios, col n)
  float sCol[64];
  {
    int n = tid & 63, ch = tid >> 6;
    const float* colp = scen + (ch * 64) * 65 + n;
#pragma unroll
    for (int s2 = 0; s2 < 64; ++s2) sCol[s2] = colp[s2 * 65];
  }

  // ---- 4) initial w0 = project(1/N) ----
  const float c = 0.5f * is_crisis[b];
  if (tid < NN) vv[tid] = (1.0f / 64.0f) - ((tid == BIL) ? c : 0.0f);
  __syncthreads();
  project_block(vv, uu, w, scal, tid, c);

  // ---- 5) 200 projected subgradient iterations ----
  for (int it = 0; it < N_ITERS; ++it) {
    // loss = -scen @ w via WMMA (w broadcast into all 16 B columns).
    // B-fragments depend only on lane half, loaded once, reused for both tiles.
    v2f wF[16];
#pragma unroll
    for (int ks = 0; ks < 16; ++ks) {
      int ka = ks * 4 + 2 * halfsel;
      v2f tw = {w[ka], w[ka + 1]};
      wF[ks] = tw;
    }
#pragma unroll
    for (int half = 0; half < 2; ++half) {
      v8f acc = {};
#pragma unroll
      for (int ks = 0; ks < 16; ++ks)
        acc = __builtin_amdgcn_wmma_f32_16x16x4_f32(
            false, aF[half][ks], false, wF[ks], (short)0, acc, false, false);
      if (mrow == 0) {  // lanes 0 and 16 hold column N=0 (rows M and M+8)
        int sbase = wv * 32 + half * 16 + 8 * halfsel;
#pragma unroll
        for (int v = 0; v < 8; ++v) loss[sbase + v] = -acc[v];
      }
    }
    __syncthreads();

    // alpha = order statistic Q_IDX via stable counting rank (thread == scenario)
    float ls = loss[tid];
    int rank = 0;
    {
      const float4* l4 = (const float4*)loss;
#pragma unroll 4
      for (int q = 0; q < 64; ++q) {
        float4 y = l4[q];
        int s2 = q * 4;
        rank += ((y.x < ls) || (y.x == ls && (s2 + 0) < tid)) ? 1 : 0;
        rank += ((y.y < ls) || (y.y == ls && (s2 + 1) < tid)) ? 1 : 0;
        rank += ((y.z < ls) || (y.z == ls && (s2 + 2) < tid)) ? 1 : 0;
        rank += ((y.w < ls) || (y.w == ls && (s2 + 3) < tid)) ? 1 : 0;
      }
    }
    if (rank == Q_IDX) scal[1] = ls;
    __syncthreads();
    const float alpha = scal[1];

    // coef_s = H(r)/(S(1-beta)) + [s==s*](1 - sumH/(S(1-beta)))
    float r = ls - alpha;
    int active = (r > 0.0f) ? 1 : 0;
    unsigned long long blt = __ballot(active);
    if (lane == 0) scal[2 + wv] = (float)__popcll(blt);
    __syncthreads();
    float sumH = 0.0f;
#pragma unroll
    for (int q = 0; q < 8; ++q) sumH += scal[2 + q];
    float cf = (float)active * INV_COEF;
    if (rank == Q_IDX) cf += 1.0f - sumH * INV_COEF;
    coef[tid] = cf;
    __syncthreads();

    // g_n = -(scen^T coef)_n ; 4 partial chunks of 64 scenarios per column
    {
      int ch = tid >> 6;
      float acc2 = 0.0f;
      const float4* cf4 = (const float4*)(coef + ch * 64);
#pragma unroll
      for (int q = 0; q < 16; ++q) {
        float4 c4 = cf4[q];
        acc2 += c4.x * sCol[4 * q + 0];
        acc2 += c4.y * sCol[4 * q + 1];
        acc2 += c4.z * sCol[4 * q + 2];
        acc2 += c4.w * sCol[4 * q + 3];
      }
      gpart[ch * 64 + (tid & 63)] = acc2;
    }
    __syncthreads();

    // w' = w - lr*g = w + lr*(scen^T coef); shift BIL by c before projection
    if (tid < NN) {
      float sc = gpart[tid] + gpart[64 + tid] + gpart[128 + tid] + gpart[192 + tid];
      float lr = 2.0f / sqrtf(1.0f + (float)it);
      float wn = w[tid] + lr * sc;
      if (tid == BIL) wn -= c;
      vv[tid] = wn;
    }
    __syncthreads();
    project_block(vv, uu, w, scal, tid, c);
  }

  // ---- 6) write result ----
  if (tid < NN) out[(size_t)b * NN + tid] = w[tid];
}

extern "C" void kernel_launch(void* const* d_in, const int* in_sizes, int n_in,
                              void* d_out, int out_size, void* d_ws, size_t ws_size,
                              hipStream_t stream) {
  (void)d_ws; (void)ws_size; (void)n_in; (void)out_size;
  const float* mu = (const float*)d_in[0];
  const float* sigma = (const float*)d_in[1];
  const float* eps = (const float*)d_in[2];
  const float* is_crisis = (const float*)d_in[3];
  float* out = (float*)d_out;

  const int B = in_sizes[3];  // is_crisis has B elements
  const size_t shmem = (size_t)LDS_FLOATS * sizeof(float);  // ~87 KB
  hipFuncSetAttribute((const void*)cvar_layer_kernel,
                      hipFuncAttributeMaxDynamicSharedMemorySize, (int)shmem);
  cvar_layer_kernel<<<dim3(B), dim3(256), shmem, stream>>>(mu, sigma, eps,
                                                           is_crisis, out);
}